// FeaturePropogation_63471026700452
// MI455X (gfx1250) — compile-verified
//
#include <hip/hip_runtime.h>

typedef float v2f __attribute__((ext_vector_type(2)));
typedef float v8f __attribute__((ext_vector_type(8)));

#define B_   4
#define N_   16384
#define M_   4096
#define C_   256
#define EPSF 1e-8f

#define NT   64        // columns (points) per workgroup in the MLP kernel
#define KC   32        // k-chunk staged in LDS
#define KW   36        // W-stage row stride (floats): 16B-aligned rows, bank-clean
#define XS   72        // x-stage row stride
#define YS   260       // y1 row stride ([n][ch] layout)

#define S_WAIT_ASYNCCNT(n) asm volatile("s_wait_asynccnt " n ::: "memory")

// one async 16B global->LDS transfer per lane (ASYNCcnt-tracked, bypasses VGPRs)
__device__ __forceinline__ void async_b128(unsigned lds_off, unsigned voff,
                                           const float* sbase) {
    asm volatile("global_load_async_to_lds_b128 %0, %1, %2"
                 :: "v"(lds_off), "v"(voff), "s"(sbase)
                 : "memory");
}

__device__ __forceinline__ unsigned lds_byte_off(const void* p) {
    // low 32 bits of the flat address of a __shared__ object == LDS byte offset
    return (unsigned)(unsigned long long)p;
}

// ---------------------------------------------------------------------------
// Kernel A: 3-NN search + inverse-distance interpolation
// ---------------------------------------------------------------------------
__global__ __launch_bounds__(256) void three_interp_kernel(
    const float* __restrict__ p1, const float* __restrict__ p2,
    const float* __restrict__ f2, float* __restrict__ interp)
{
    __shared__ float p2s[M_ * 3];
    const int b = blockIdx.y;
    const int n = blockIdx.x * 256 + threadIdx.x;

    const float* p2b = p2 + (size_t)b * M_ * 3;
    for (int i = threadIdx.x; i < M_ * 3; i += 256) p2s[i] = p2b[i];
    __syncthreads();

    const size_t pbase = ((size_t)b * N_ + n) * 3;
    const float px = p1[pbase + 0];
    const float py = p1[pbase + 1];
    const float pz = p1[pbase + 2];

    float d0 = 1e30f, d1 = 1e30f, d2 = 1e30f;
    int   i0 = 0,     i1 = 0,     i2 = 0;

    #pragma unroll 4
    for (int m = 0; m < M_; ++m) {
        const float dx = px - p2s[3 * m + 0];
        const float dy = py - p2s[3 * m + 1];
        const float dz = pz - p2s[3 * m + 2];
        const float d  = dx * dx + dy * dy + dz * dz;
        if (d < d0)      { d2 = d1; i2 = i1; d1 = d0; i1 = i0; d0 = d; i0 = m; }
        else if (d < d1) { d2 = d1; i2 = i1; d1 = d;  i1 = m; }
        else if (d < d2) { d2 = d;  i2 = m; }
    }

    const float r0 = 1.0f / (d0 + EPSF);
    const float r1 = 1.0f / (d1 + EPSF);
    const float r2 = 1.0f / (d2 + EPSF);
    const float rs = 1.0f / (r0 + r1 + r2);
    const float w0 = r0 * rs, w1 = r1 * rs, w2 = r2 * rs;

    const float* f2b = f2 + (size_t)b * C_ * M_;
    float* ob = interp + (size_t)b * C_ * N_ + n;
    const float* g0 = f2b + i0;
    const float* g1 = f2b + i1;
    const float* g2 = f2b + i2;
    #pragma unroll 4
    for (int c = 0; c < C_; ++c) {
        const size_t off = (size_t)c * M_;
        ob[(size_t)c * N_] = w0 * g0[off] + w1 * g1[off] + w2 * g2[off];
    }
}

// ---------------------------------------------------------------------------
// Kernel B: fused  y1 = relu(W1 @ [f1; interp] + b1) ; out = relu(W2 @ y1 + b2)
// Double-buffered async global->LDS staging overlapped with f32 WMMA.
// ---------------------------------------------------------------------------

// stage a 256 x 32 weight chunk (row stride rowlen) into dst[o*KW + kk]
__device__ __forceinline__ void issue_w_chunk(float* dst, const float* Wg,
                                              int rowlen, int kc, int tid) {
    const unsigned base = lds_byte_off(dst);
    #pragma unroll
    for (int i = 0; i < 8; ++i) {
        const int idx = tid + i * 256;          // 0..2047
        const int o   = idx >> 3;
        const int k4  = (idx & 7) * 4;
        async_b128(base + (unsigned)((o * KW + k4) * 4),
                   (unsigned)((o * rowlen + kc + k4) * 4), Wg);
    }
}

// stage a 32 x 64 input chunk (rows of src, cols n0..n0+63) into dst[kk*XS + n]
__device__ __forceinline__ void issue_x_chunk(float* dst, const float* src,
                                              int n0, int tid) {
    const unsigned base = lds_byte_off(dst);
    #pragma unroll
    for (int i = 0; i < 2; ++i) {
        const int idx = tid + i * 256;          // 0..511
        const int kk  = idx >> 4;
        const int n4  = (idx & 15) * 4;
        async_b128(base + (unsigned)((kk * XS + n4) * 4),
                   (unsigned)((kk * N_ + n0 + n4) * 4), src);
    }
}

__global__ __launch_bounds__(256) void fp_mlp_kernel(
    const float* __restrict__ f1, const float* __restrict__ interp,
    const float* __restrict__ W1, const float* __restrict__ b1,
    const float* __restrict__ W2, const float* __restrict__ b2,
    float* __restrict__ out)
{
    __shared__ __align__(16) float wl[2][C_ * KW];   // 2 x 36864 B
    __shared__ __align__(16) float xl[2][KC * XS];   // 2 x  9216 B
    __shared__ __align__(16) float y1[NT * YS];      //     66560 B  ([n][ch])

    const int tid  = threadIdx.x;
    const int lane = tid & 31;
    const int wv   = tid >> 5;        // wave 0..7
    const int half = lane >> 4;
    const int l15  = lane & 15;
    const int b    = blockIdx.y;
    const int n0   = blockIdx.x * NT;
    const int nt   = wv & 3;          // n-subtile (16 cols)
    const int oh   = wv >> 2;         // 0..1 -> owns o-tiles [oh*8, oh*8+8)
    const int col  = nt * 16 + l15;

    const float* f1b = f1     + (size_t)b * C_ * N_;
    const float* inb = interp + (size_t)b * C_ * N_;

    const v8f zero = {};

    // ---------------- GEMM1: 16 chunks of K=32 over 2C=512 ----------------
    v8f acc[8];
    #pragma unroll
    for (int t = 0; t < 8; ++t) acc[t] = zero;

    // prologue: prefetch chunk 0
    issue_w_chunk(wl[0], W1, 2 * C_, 0, tid);
    issue_x_chunk(xl[0], f1b, n0, tid);

    int cur = 0;
    for (int ch = 0; ch < 16; ++ch) {
        const int kcn = (ch + 1) * KC;
        if (ch + 1 < 16) {
            issue_w_chunk(wl[cur ^ 1], W1, 2 * C_, kcn, tid);
            const float* src = (kcn < C_) ? (f1b + (size_t)kcn * N_)
                                          : (inb + (size_t)(kcn - C_) * N_);
            issue_x_chunk(xl[cur ^ 1], src, n0, tid);
            S_WAIT_ASYNCCNT("10");        // leave only next chunk outstanding
        } else {
            S_WAIT_ASYNCCNT("0");
        }
        __syncthreads();

        const float* wl_ = wl[cur];
        const float* xl_ = xl[cur];
        v2f bf[8];
        #pragma unroll
        for (int ks = 0; ks < 8; ++ks) {
            const int r = ks * 4 + half * 2;
            bf[ks].x = xl_[r * XS + col];
            bf[ks].y = xl_[(r + 1) * XS + col];
        }
        #pragma unroll
        for (int t = 0; t < 8; ++t) {
            const int orow = ((oh * 8 + t) * 16 + l15) * KW;
            #pragma unroll
            for (int ks = 0; ks < 8; ++ks) {
                const v2f af = *(const v2f*)&wl_[orow + ks * 4 + 2 * half];
                acc[t] = __builtin_amdgcn_wmma_f32_16x16x4_f32(
                    false, af, false, bf[ks], (short)0, acc[t], false, false);
            }
        }
        __syncthreads();
        cur ^= 1;
    }

    // y1 tile (bias + relu) into LDS, [n][ch] layout (contiguous 8 per lane)
    #pragma unroll
    for (int t = 0; t < 8; ++t) {
        const int ob = (oh * 8 + t) * 16 + 8 * half;
        #pragma unroll
        for (int v = 0; v < 8; ++v) {
            const int o = ob + v;
            y1[col * YS + o] = fmaxf(acc[t][v] + b1[o], 0.0f);
        }
    }
    // prefetch W2 chunk 0 (wl[cur] was fully consumed before last barrier)
    issue_w_chunk(wl[cur], W2, C_, 0, tid);
    __syncthreads();

    // ---------------- GEMM2: 8 chunks of K=32 over C=256 ----------------
    v8f acc2[8];
    #pragma unroll
    for (int t = 0; t < 8; ++t) acc2[t] = zero;

    for (int ch = 0; ch < 8; ++ch) {
        if (ch + 1 < 8) {
            issue_w_chunk(wl[cur ^ 1], W2, C_, (ch + 1) * KC, tid);
            S_WAIT_ASYNCCNT("8");
        } else {
            S_WAIT_ASYNCCNT("0");
        }
        __syncthreads();

        const float* wl_ = wl[cur];
        v2f bf[8];
        #pragma unroll
        for (int ks = 0; ks < 8; ++ks)
            bf[ks] = *(const v2f*)&y1[col * YS + ch * KC + ks * 4 + 2 * half];
        #pragma unroll
        for (int t = 0; t < 8; ++t) {
            const int orow = ((oh * 8 + t) * 16 + l15) * KW;
            #pragma unroll
            for (int ks = 0; ks < 8; ++ks) {
                const v2f af = *(const v2f*)&wl_[orow + ks * 4 + 2 * half];
                acc2[t] = __builtin_amdgcn_wmma_f32_16x16x4_f32(
                    false, af, false, bf[ks], (short)0, acc2[t], false, false);
            }
        }
        __syncthreads();
        cur ^= 1;
    }

    // store output (bias + relu), coalesced per half-wave
    float* outb = out + (size_t)b * C_ * N_;
    #pragma unroll
    for (int t = 0; t < 8; ++t) {
        const int ob = (oh * 8 + t) * 16 + 8 * half;
        #pragma unroll
        for (int v = 0; v < 8; ++v) {
            const int o = ob + v;
            outb[(size_t)o * N_ + n0 + col] = fmaxf(acc2[t][v] + b2[o], 0.0f);
        }
    }
}

// ---------------------------------------------------------------------------
extern "C" void kernel_launch(void* const* d_in, const int* in_sizes, int n_in,
                              void* d_out, int out_size, void* d_ws, size_t ws_size,
                              hipStream_t stream) {
    const float* p1 = (const float*)d_in[0];
    const float* f1 = (const float*)d_in[1];
    const float* p2 = (const float*)d_in[2];
    const float* f2 = (const float*)d_in[3];
    const float* W1 = (const float*)d_in[4];
    const float* b1 = (const float*)d_in[5];
    const float* W2 = (const float*)d_in[6];
    const float* b2 = (const float*)d_in[7];
    float* out    = (float*)d_out;
    float* interp = (float*)d_ws;   // B*C*N floats = 64 MiB scratch

    dim3 gA(N_ / 256, B_);
    three_interp_kernel<<<gA, 256, 0, stream>>>(p1, p2, f2, interp);

    dim3 gB(N_ / NT, B_);
    fp_mlp_kernel<<<gB, 256, 0, stream>>>(f1, interp, W1, b1, W2, b2, out);
}